// Lp_Loss_70815420776931
// MI455X (gfx1250) — compile-verified
//
#include <hip/hip_runtime.h>
#include <hip/hip_bf16.h>

// Problem constants (from reference)
#define NJ      17          // NUM_JOINTS
#define NB      16          // batch
#define NP      30          // persons
#define HW      65536       // 256*256
#define CPB     34          // channels in y_pred (2*NJ)
#define BLKS_B  64          // blocks per batch for heatmap pass
#define TPB     256         // threads per block (8 waves on wave32)

typedef __attribute__((ext_vector_type(2))) float v2f;
typedef __attribute__((ext_vector_type(4))) float v4f;
typedef __attribute__((ext_vector_type(8))) float v8f;

// Full 32-lane f32 sum using V_WMMA_F32_16X16X4_F32.
// Step 1: A = per-lane value in K=0/K=2 slots (vgpr1 = 0), B = ones
//         -> D1[m][n] = x[m] + x[m+16]  (row-sum broadcast across columns)
// Step 2: 4 accumulating WMMAs, A = ones, B = (d1[2i], d1[2i+1])
//         -> every element of C = sum of all 16 row-sums = total of 32 lanes.
// Exact (f32 accumulate, RNE), deterministic, EXEC all-ones required (callers
// invoke it from fully-active blocks).
__device__ __forceinline__ float wave_reduce_wmma(float x) {
    v2f a;    a.x = x;     a.y = 0.0f;
    v2f ones; ones.x = 1.0f; ones.y = 1.0f;
    v8f zero = {};
    v8f d1 = __builtin_amdgcn_wmma_f32_16x16x4_f32(
        false, a, false, ones, (short)0, zero, false, false);
    v8f c = zero;
#pragma unroll
    for (int i = 0; i < 4; ++i) {
        v2f bp; bp.x = d1[2 * i]; bp.y = d1[2 * i + 1];
        c = __builtin_amdgcn_wmma_f32_16x16x4_f32(
            false, ones, false, bp, (short)0, c, false, false);
    }
    return c[0];
}

// ---------------------------------------------------------------------------
// Pass 1: masked-MSE partial sums. One batch is covered by BLKS_B blocks;
// each thread does exactly 17 float4 iterations (278528 float4 / 16384 thr).
// y_pred / gt_heatmap are streamed once -> non-temporal loads (TH_NT) keep
// the 192MB L2 clean; gt_mask is reused (64 blocks x 17 joints per batch)
// so it stays on the default temporal path.
// ---------------------------------------------------------------------------
__global__ void heat_partial(const float* __restrict__ y_pred,
                             const float* __restrict__ gt_hm,
                             const float* __restrict__ gt_mask,
                             float* __restrict__ ws_heat) {
    const int b   = blockIdx.x / BLKS_B;
    const int blk = blockIdx.x % BLKS_B;
    const int t   = blk * TPB + threadIdx.x;          // [0, 16384)

    const v4f* __restrict__ p4 = (const v4f*)(y_pred  + (size_t)b * CPB * HW);
    const v4f* __restrict__ g4 = (const v4f*)(gt_hm   + (size_t)b * NJ  * HW);
    const v4f* __restrict__ m4 = (const v4f*)(gt_mask + (size_t)b * HW);

    const int N4     = NJ * HW / 4;                   // 278528
    const int STRIDE = BLKS_B * TPB;                  // 16384

    float acc = 0.0f;
    for (int q = t; q < N4; q += STRIDE) {
        v4f p = __builtin_nontemporal_load(&p4[q]);
        v4f g = __builtin_nontemporal_load(&g4[q]);
        v4f m = m4[q & (HW / 4 - 1)];                 // pixel4 = q % 16384
        float dx = p.x - g.x, dy = p.y - g.y, dz = p.z - g.z, dw = p.w - g.w;
        acc = fmaf(dx * dx, m.x, acc);
        acc = fmaf(dy * dy, m.y, acc);
        acc = fmaf(dz * dz, m.z, acc);
        acc = fmaf(dw * dw, m.w, acc);
    }

    float wsum = wave_reduce_wmma(acc);

    __shared__ float lds[TPB / 32];
    const int wid = threadIdx.x >> 5;
    if ((threadIdx.x & 31) == 0) lds[wid] = wsum;
    __syncthreads();
    if (threadIdx.x == 0) {
        float s = 0.0f;
#pragma unroll
        for (int w = 0; w < TPB / 32; ++w) s += lds[w];
        ws_heat[blockIdx.x] = s;
    }
}

// ---------------------------------------------------------------------------
// Tag loss, per batch: one block of 32 lanes, one person per lane (30 used).
// Gathers tag values from the second half of y_pred; computes pull + push.
// ---------------------------------------------------------------------------
__global__ void tag_batch(const float* __restrict__ y_pred,
                          const int* __restrict__ joints,
                          float* __restrict__ ws_tag) {
    const int b = blockIdx.x;
    const float* __restrict__ tag_map =
        y_pred + (size_t)b * CPB * HW + (size_t)NJ * HW;
    const int* __restrict__ jb = joints + b * NP * NJ * 2;
    const int p = threadIdx.x;

    __shared__ float s_tm[NP], s_has[NP], s_pull[NP], s_row[NP];

    float tm = 0.0f, haspf = 0.0f, pullp = 0.0f;
    if (p < NP) {
        float tags[NJ], vis[NJ];
        float pjc = 0.0f, tsum = 0.0f;
#pragma unroll
        for (int j = 0; j < NJ; ++j) {
            int   loc = jb[(p * NJ + j) * 2 + 0];
            float v   = (float)jb[(p * NJ + j) * 2 + 1];
            float tg  = tag_map[loc] * v;
            tags[j] = tg; vis[j] = v;
            pjc += v; tsum += tg;
        }
        float pjc_safe = (pjc == 0.0f) ? 1.0f : pjc;
        tm = tsum / pjc_safe;
        float pl = 0.0f;
#pragma unroll
        for (int j = 0; j < NJ; ++j) {
            float d = tags[j] - tm;
            pl = fmaf(vis[j] * d, d, pl);
        }
        pl /= pjc_safe;
        pullp = (pjc == 0.0f) ? 0.0f : pl;
        haspf = (pjc > 0.0f) ? 1.0f : 0.0f;
    }
    if (p < NP) { s_tm[p] = tm; s_has[p] = haspf; s_pull[p] = pullp; }
    __syncthreads();

    if (p < NP) {
        float r = 0.0f;
        for (int q = 0; q < NP; ++q) {
            float d = s_tm[p] - s_tm[q];
            r = fmaf(__expf(-d * d), s_has[q], r);
        }
        s_row[p] = r * s_has[p];
    }
    __syncthreads();

    if (p == 0) {
        float pc = 0.0f, pulls = 0.0f, pushs = 0.0f;
        for (int q = 0; q < NP; ++q) {
            pc += s_has[q]; pulls += s_pull[q]; pushs += s_row[q];
        }
        float pcnt   = (pc == 0.0f) ? 1.0f : pc;
        float pull_b = pulls / pcnt;
        float denom  = fmaxf((pc - 1.0f) * pc, 1.0f);
        float push_b = 0.5f * (pushs - pc) / denom;
        if (pc < 2.0f) push_b = 0.0f;
        ws_tag[b] = pull_b + push_b;
    }
}

// ---------------------------------------------------------------------------
// Finalize: blocks 0..15 reduce heatmap partials (/= J*H*W); block 16 averages
// the per-batch tag terms and applies TAG_LOSS_WEIGHT.
// ---------------------------------------------------------------------------
__global__ void finalize(const float* __restrict__ ws_heat,
                         const float* __restrict__ ws_tag,
                         float* __restrict__ out) {
    const int blk = blockIdx.x;
    if (blk < NB) {
        __shared__ float lds[BLKS_B];
        lds[threadIdx.x] = ws_heat[blk * BLKS_B + threadIdx.x];
        __syncthreads();
        for (int s = BLKS_B / 2; s > 0; s >>= 1) {
            if ((int)threadIdx.x < s) lds[threadIdx.x] += lds[threadIdx.x + s];
            __syncthreads();
        }
        if (threadIdx.x == 0)
            out[blk] = lds[0] / (float)(NJ * HW);
    } else if (threadIdx.x == 0) {
        float s = 0.0f;
        for (int i = 0; i < NB; ++i) s += ws_tag[i];
        out[NB] = (s / (float)NB) * 0.001f;
    }
}

extern "C" void kernel_launch(void* const* d_in, const int* in_sizes, int n_in,
                              void* d_out, int out_size, void* d_ws, size_t ws_size,
                              hipStream_t stream) {
    const float* y_pred    = (const float*)d_in[0];
    const float* gt_hm     = (const float*)d_in[1];
    const float* gt_mask   = (const float*)d_in[2];
    const int*   gt_joints = (const int*)d_in[3];
    float*       out       = (float*)d_out;

    float* ws_heat = (float*)d_ws;                 // NB*BLKS_B floats
    float* ws_tag  = ws_heat + NB * BLKS_B;        // NB floats

    heat_partial<<<NB * BLKS_B, TPB, 0, stream>>>(y_pred, gt_hm, gt_mask, ws_heat);
    tag_batch<<<NB, 32, 0, stream>>>(y_pred, gt_joints, ws_tag);
    finalize<<<NB + 1, BLKS_B, 0, stream>>>(ws_heat, ws_tag, out);
}